// BidirMamba_1400159338678
// MI455X (gfx1250) — compile-verified
//
#include <hip/hip_runtime.h>

// ---------------- problem constants ----------------
#define L_SEQ   8192          // 2 * 64 * 64 tokens
#define DMODEL  128
#define DINNER  256
#define DSTATE  16
#define DTRANK  8
#define NXDBL   40            // dt(8) + B(16) + C(16)
#define NCHUNK  64
#define CSIZE   128           // NCHUNK * CSIZE == L_SEQ
#define HW      4096          // 64*64
#define NBATCH  2

typedef float v2f __attribute__((ext_vector_type(2)));
typedef float v8f __attribute__((ext_vector_type(8)));
typedef unsigned int u32x4 __attribute__((ext_vector_type(4)));
typedef int i32x4 __attribute__((ext_vector_type(4)));
typedef int i32x8 __attribute__((ext_vector_type(8)));

__device__ __forceinline__ float silu_f(float x) {
    return x * (1.0f / (1.0f + __expf(-x)));
}
__device__ __forceinline__ float softplus_f(float x) {
    return (x > 20.0f) ? x : __logf(1.0f + __expf(x));
}

// ---- Tensor Data Mover: async 1-D copy of nelem f32 from global -> LDS ---------
// D# per CDNA5 ISA ch.8: group0 = {count, lds_addr, global_addr, type=2("image")},
// group1 = {data_size=4B, tensor_dim0 = nelem, tile_dim0 = nelem, dim1 = 1}.
__device__ __forceinline__ void tdm_load_1d(void* lds_dst, const void* gsrc,
                                            unsigned nelem) {
    unsigned lds_off = (unsigned)(unsigned long long)lds_dst;   // LDS aperture: low 32b = offset
    unsigned long long ga = (unsigned long long)gsrc;
    u32x4 g0;
    g0[0] = 1u;                                              // count=1, user descriptor
    g0[1] = lds_off;                                         // lds_addr (bytes)
    g0[2] = (unsigned)(ga & 0xFFFFFFFFull);                  // global_addr[31:0]
    g0[3] = ((unsigned)((ga >> 32) & 0x01FFFFFFull))         // global_addr[56:32]
            | 0x80000000u;                                   // type=2 -> bits[127:126]=10b
    i32x8 g1;
    g1[0] = (int)(2u << 16);                                 // data_size = 2 (4 bytes)
    g1[1] = (int)((nelem & 0xFFFFu) << 16);                  // tensor_dim0[15:0]
    g1[2] = (int)((nelem >> 16) & 0xFFFFu) | (int)(1u << 16);// tensor_dim0[31:16] | tensor_dim1=1
    g1[3] = (int)((nelem & 0xFFFFu) << 16);                  // tile_dim0 = nelem (<=65535)
    g1[4] = 1;                                               // tile_dim1 = 1, tile_dim2 = 0
    g1[5] = (int)nelem;                                      // tensor_dim0_stride[31:0]
    g1[6] = 0;                                               // stride hi / dim1_stride lo
    g1[7] = 0;
    i32x4 z4 = {0, 0, 0, 0};
#if __clang_major__ >= 23
    i32x8 z8 = {0, 0, 0, 0, 0, 0, 0, 0};
    __builtin_amdgcn_tensor_load_to_lds(g0, g1, z4, z4, z8, 0);
#else
    __builtin_amdgcn_tensor_load_to_lds(g0, g1, z4, z4, 0);
#endif
}

// ---------------- 0: x (B,C,H,W) -> xT (L, DMODEL) ----------------
__global__ void bm_transpose(const float* __restrict__ x, float* __restrict__ xT) {
    int i = blockIdx.x * blockDim.x + threadIdx.x;   // over 2*128*4096 = 2^20
    if (i >= NBATCH * DMODEL * HW) return;
    int s = i & (HW - 1);
    int c = (i >> 12) & (DMODEL - 1);
    int b = i >> 19;
    xT[(b * HW + s) * DMODEL + c] = x[i];
}

// ------- 1: xz(L,512) = xT(L,128) @ Win(512,128)^T  (TDM->LDS + WMMA f32) -------
// Block = 8 waves: one 16-col strip of Win (nt), 8 consecutive M tiles (mg).
__global__ void bm_gemm_xz(const float* __restrict__ A, const float* __restrict__ W,
                           float* __restrict__ C) {
    __shared__ float sA[128 * DMODEL];   // 8 M-tiles x 16 rows x K=128  (64 KB)
    __shared__ float sB[16 * DMODEL];    // 16 N-rows x K=128            (8 KB)
    int nt = blockIdx.x & 31;            // 32 N strips
    int mg = blockIdx.x >> 5;            // 64 M groups of 8 tiles
    if (threadIdx.x < 32) {              // wave 0 drives the DMA
        tdm_load_1d(sA, A + (size_t)mg * 128 * DMODEL, 128 * DMODEL);
        tdm_load_1d(sB, W + (size_t)nt * 16 * DMODEL, 16 * DMODEL);
        __builtin_amdgcn_s_wait_tensorcnt((short)0);
    }
    __syncthreads();
    int wave = threadIdx.x >> 5, lane = threadIdx.x & 31;
    int rowA  = wave * 16 + (lane & 15);
    int colB  = lane & 15;
    int khalf = (lane >> 4) * 2;
    v8f acc = {};
    for (int kb = 0; kb < DMODEL; kb += 4) {
        v2f a, b;
        a.x = sA[rowA * DMODEL + kb + khalf + 0];
        a.y = sA[rowA * DMODEL + kb + khalf + 1];
        b.x = sB[colB * DMODEL + kb + khalf + 0];
        b.y = sB[colB * DMODEL + kb + khalf + 1];
        acc = __builtin_amdgcn_wmma_f32_16x16x4_f32(false, a, false, b,
                                                    (short)0, acc, false, false);
    }
    int crow = (mg * 8 + wave) * 16 + 8 * (lane >> 4);
    int ccol = nt * 16 + (lane & 15);
#pragma unroll
    for (int r = 0; r < 8; r++) C[(crow + r) * 512 + ccol] = acc[r];
}

// ---------------- 2: causal depthwise conv(k=4) + bias + SiLU, per direction ----
__global__ void bm_conv(const float* __restrict__ xz, const float* __restrict__ cw,
                        const float* __restrict__ cb, float* __restrict__ U) {
    int i = blockIdx.x * blockDim.x + threadIdx.x;   // 2 * L * 256 = 2^22
    if (i >= 2 * L_SEQ * DINNER) return;
    int e   = i & (DINNER - 1);
    int l   = (i >> 8) & (L_SEQ - 1);
    int dir = i >> 21;
    float acc = cb[e];
#pragma unroll
    for (int k = 0; k < 4; k++) {
        int ls = l - 3 + k;
        if (ls >= 0) {
            int pos = dir ? (L_SEQ - 1 - ls) : ls;     // sequence is flipped for bwd
            acc += cw[e * 4 + k] * xz[pos * 512 + e];  // u_pre = xz[:, 0:256]
        }
    }
    U[i] = silu_f(acc);
}

// ---------------- 3: x_dbl(L,40) = U(L,256) @ Wx(40,256)^T (WMMA f32, N pad 48) --
__global__ void bm_gemm_xdbl(const float* __restrict__ U, const float* __restrict__ Wx,
                             float* __restrict__ XD) {
    const int NT = 3;                                  // 48 padded cols
    int wave = (blockIdx.x * blockDim.x + threadIdx.x) >> 5;
    int lane = threadIdx.x & 31;
    int dir  = wave / ((L_SEQ / 16) * NT);
    int rem  = wave % ((L_SEQ / 16) * NT);
    if (dir >= 2) return;
    int mt = rem / NT, nt = rem % NT;
    const float* A = U  + (size_t)dir * L_SEQ * DINNER;
    float*       C = XD + (size_t)dir * L_SEQ * NXDBL;
    int rowA  = mt * 16 + (lane & 15);
    int colB  = nt * 16 + (lane & 15);
    int khalf = (lane >> 4) * 2;
    bool colOK = (colB < NXDBL);
    v8f acc = {};
    for (int kb = 0; kb < DINNER; kb += 4) {
        v2f a, b;
        a.x = A[rowA * DINNER + kb + khalf + 0];
        a.y = A[rowA * DINNER + kb + khalf + 1];
        b.x = colOK ? Wx[colB * DINNER + kb + khalf + 0] : 0.0f;
        b.y = colOK ? Wx[colB * DINNER + kb + khalf + 1] : 0.0f;
        acc = __builtin_amdgcn_wmma_f32_16x16x4_f32(false, a, false, b,
                                                    (short)0, acc, false, false);
    }
    if (colOK) {
        int crow = mt * 16 + 8 * (lane >> 4);
#pragma unroll
        for (int r = 0; r < 8; r++) C[(crow + r) * NXDBL + colB] = acc[r];
    }
}

// ---------------- 4: delta = softplus(dt @ Wdt^T + bdt) --------------------------
__global__ void bm_delta(const float* __restrict__ XD, const float* __restrict__ Wdt,
                         const float* __restrict__ bdt, float* __restrict__ DL) {
    int i = blockIdx.x * blockDim.x + threadIdx.x;   // 2 * L * 256
    if (i >= 2 * L_SEQ * DINNER) return;
    int e   = i & (DINNER - 1);
    int l   = (i >> 8) & (L_SEQ - 1);
    int dir = i >> 21;
    const float* xd = XD + ((size_t)dir * L_SEQ + l) * NXDBL;
    float acc = bdt[e];
#pragma unroll
    for (int r = 0; r < DTRANK; r++) acc += xd[r] * Wdt[e * DTRANK + r];
    DL[i] = softplus_f(acc);
}

// ---------------- 5: scan phase 1 — per-chunk (prod dA, local end state) ---------
__global__ void bm_scan1(const float* __restrict__ DL, const float* __restrict__ U,
                         const float* __restrict__ XD, const float* __restrict__ A_log,
                         float* __restrict__ P, float* __restrict__ Hc) {
    int blk = blockIdx.x;              // dir*NCHUNK + chunk
    int dir = blk >> 6, c = blk & 63;
    int e = threadIdx.x;               // channel, 256 threads
    float Areg[DSTATE];
#pragma unroll
    for (int n = 0; n < DSTATE; n++) Areg[n] = -__expf(A_log[e * DSTATE + n]);
    float p[DSTATE], h[DSTATE];
#pragma unroll
    for (int n = 0; n < DSTATE; n++) { p[n] = 1.0f; h[n] = 0.0f; }
    size_t base = (size_t)dir * L_SEQ;
    for (int i = 0; i < CSIZE; i++) {
        int l = c * CSIZE + i;
        __builtin_prefetch(DL + (base + l + 8) * DINNER + e, 0, 1);
        __builtin_prefetch(U  + (base + l + 8) * DINNER + e, 0, 1);
        float d  = DL[(base + l) * DINNER + e];
        float uu = U [(base + l) * DINNER + e];
        const float* Bv = XD + (base + l) * NXDBL + DTRANK;
        float du = d * uu;
#pragma unroll
        for (int n = 0; n < DSTATE; n++) {
            float dA = __expf(d * Areg[n]);
            p[n] *= dA;
            h[n] = dA * h[n] + du * Bv[n];
        }
    }
    size_t o = ((size_t)blk * DINNER + e) * DSTATE;
#pragma unroll
    for (int n = 0; n < DSTATE; n++) { P[o + n] = p[n]; Hc[o + n] = h[n]; }
}

// ---------------- 6: scan phase 2 — sequential combine over 64 chunks ------------
__global__ void bm_scan2(const float* __restrict__ P, const float* __restrict__ Hc,
                         float* __restrict__ Hin) {
    int idx = blockIdx.x * blockDim.x + threadIdx.x;  // 2 * 256*16 = 8192
    if (idx >= 2 * DINNER * DSTATE) return;
    int dir = idx >> 12;
    int en  = idx & (DINNER * DSTATE - 1);
    const size_t stride = (size_t)DINNER * DSTATE;    // per chunk
    size_t b0 = (size_t)dir * NCHUNK * stride + en;
    float h = 0.0f;
    for (int c = 0; c < NCHUNK; c++) {
        Hin[b0 + c * stride] = h;
        h = P[b0 + c * stride] * h + Hc[b0 + c * stride];
    }
}

// ---------------- 7: scan phase 3 — replay with prefix, fuse +u*D and *silu(z) ---
__global__ void bm_scan3(const float* __restrict__ DL, const float* __restrict__ U,
                         const float* __restrict__ XD, const float* __restrict__ A_log,
                         const float* __restrict__ Hin, const float* __restrict__ xz,
                         const float* __restrict__ Dv, float* __restrict__ Y) {
    int blk = blockIdx.x;
    int dir = blk >> 6, c = blk & 63;
    int e = threadIdx.x;
    float Areg[DSTATE];
#pragma unroll
    for (int n = 0; n < DSTATE; n++) Areg[n] = -__expf(A_log[e * DSTATE + n]);
    float h[DSTATE];
    size_t o = ((size_t)blk * DINNER + e) * DSTATE;
#pragma unroll
    for (int n = 0; n < DSTATE; n++) h[n] = Hin[o + n];
    float De = Dv[e];
    size_t base = (size_t)dir * L_SEQ;
    for (int i = 0; i < CSIZE; i++) {
        int l = c * CSIZE + i;
        __builtin_prefetch(DL + (base + l + 8) * DINNER + e, 0, 1);
        __builtin_prefetch(U  + (base + l + 8) * DINNER + e, 0, 1);
        float d  = DL[(base + l) * DINNER + e];
        float uu = U [(base + l) * DINNER + e];
        const float* Bv = XD + (base + l) * NXDBL + DTRANK;
        const float* Cv = XD + (base + l) * NXDBL + DTRANK + DSTATE;
        float du = d * uu;
        float y = 0.0f;
#pragma unroll
        for (int n = 0; n < DSTATE; n++) {
            float dA = __expf(d * Areg[n]);
            h[n] = dA * h[n] + du * Bv[n];
            y += h[n] * Cv[n];
        }
        y += uu * De;
        int orig = dir ? (L_SEQ - 1 - l) : l;          // map back to original order
        float z = xz[orig * 512 + DINNER + e];         // z = xz[:, 256:512]
        Y[(base + orig) * DINNER + e] = y * silu_f(z);
    }
}

// ------- 8: ym(L,128) = (Yfwd+Ybwd)(L,256) @ Wout(128,256)^T (TDM B + WMMA) ------
__global__ void bm_gemm_out(const float* __restrict__ Y, const float* __restrict__ W,
                            float* __restrict__ C) {
    __shared__ float sB[16 * DINNER];    // one 16-row strip of Wout (16 KB)
    int nt = blockIdx.x & 7;             // 8 N strips
    int mg = blockIdx.x >> 3;            // 64 M groups of 8 tiles
    if (threadIdx.x < 32) {
        tdm_load_1d(sB, W + (size_t)nt * 16 * DINNER, 16 * DINNER);
        __builtin_amdgcn_s_wait_tensorcnt((short)0);
    }
    __syncthreads();
    int wave = threadIdx.x >> 5, lane = threadIdx.x & 31;
    const size_t Loff = (size_t)L_SEQ * DINNER;         // second direction
    int rowA  = (mg * 8 + wave) * 16 + (lane & 15);
    int colB  = lane & 15;
    int khalf = (lane >> 4) * 2;
    v8f acc = {};
    for (int kb = 0; kb < DINNER; kb += 4) {
        v2f a, b;
        size_t ia = (size_t)rowA * DINNER + kb + khalf;
        a.x = Y[ia + 0] + Y[Loff + ia + 0];
        a.y = Y[ia + 1] + Y[Loff + ia + 1];
        b.x = sB[colB * DINNER + kb + khalf + 0];
        b.y = sB[colB * DINNER + kb + khalf + 1];
        acc = __builtin_amdgcn_wmma_f32_16x16x4_f32(false, a, false, b,
                                                    (short)0, acc, false, false);
    }
    int crow = (mg * 8 + wave) * 16 + 8 * (lane >> 4);
    int ccol = nt * 16 + (lane & 15);
#pragma unroll
    for (int r = 0; r < 8; r++) C[(crow + r) * DMODEL + ccol] = acc[r];
}

// ---------------- 9: GroupNorm statistics (8 blocks: batch x group) --------------
__global__ void bm_gnstats(const float* __restrict__ Ym, float* __restrict__ stats) {
    int b = blockIdx.x >> 2, g = blockIdx.x & 3;
    float s = 0.0f, s2 = 0.0f;
    for (int t = threadIdx.x; t < HW * 32; t += 256) {
        int li = t >> 5, ci = t & 31;
        float v = Ym[(size_t)(b * HW + li) * DMODEL + g * 32 + ci];
        s += v; s2 += v * v;
    }
    __shared__ float sh[256], sh2[256];
    sh[threadIdx.x] = s; sh2[threadIdx.x] = s2;
    __syncthreads();
    for (int off = 128; off > 0; off >>= 1) {
        if ((int)threadIdx.x < off) {
            sh[threadIdx.x]  += sh[threadIdx.x + off];
            sh2[threadIdx.x] += sh2[threadIdx.x + off];
        }
        __syncthreads();
    }
    if (threadIdx.x == 0) {
        const float N = (float)(HW * 32);
        float mu  = sh[0] / N;
        float var = sh2[0] / N - mu * mu;
        stats[blockIdx.x * 2 + 0] = mu;
        stats[blockIdx.x * 2 + 1] = rsqrtf(var + 1e-5f);
    }
}

// ---------------- 10: normalize + affine + SiLU + residual -----------------------
__global__ void bm_final(const float* __restrict__ Ym, const float* __restrict__ x,
                         const float* __restrict__ gw, const float* __restrict__ gb,
                         const float* __restrict__ stats, float* __restrict__ out) {
    int i = blockIdx.x * blockDim.x + threadIdx.x;    // 2*128*4096
    if (i >= NBATCH * DMODEL * HW) return;
    int s = i & (HW - 1);
    int c = (i >> 12) & (DMODEL - 1);
    int b = i >> 19;
    float v  = Ym[(size_t)(b * HW + s) * DMODEL + c];
    int g = c >> 5;
    float mu = stats[(b * 4 + g) * 2 + 0];
    float rs = stats[(b * 4 + g) * 2 + 1];
    float xn = (v - mu) * rs * gw[c] + gb[c];
    out[i] = silu_f(xn) + x[i];
}

// ---------------- host-side orchestration ---------------------------------------
extern "C" void kernel_launch(void* const* d_in, const int* in_sizes, int n_in,
                              void* d_out, int out_size, void* d_ws, size_t ws_size,
                              hipStream_t stream) {
    (void)in_sizes; (void)n_in; (void)out_size; (void)ws_size;
    const float* x      = (const float*)d_in[0];
    const float* Win    = (const float*)d_in[1];
    const float* conv_w = (const float*)d_in[2];
    const float* conv_b = (const float*)d_in[3];
    const float* Wx     = (const float*)d_in[4];
    const float* Wdt    = (const float*)d_in[5];
    const float* bdt    = (const float*)d_in[6];
    const float* A_log  = (const float*)d_in[7];
    const float* Dv     = (const float*)d_in[8];
    const float* Wout   = (const float*)d_in[9];
    const float* gn_w   = (const float*)d_in[10];
    const float* gn_b   = (const float*)d_in[11];
    float* out = (float*)d_out;

    float* ws = (float*)d_ws;
    size_t o = 0;
    float* xT    = ws + o; o += (size_t)L_SEQ * DMODEL;            // 1.05M
    float* xz    = ws + o; o += (size_t)L_SEQ * 512;               // 4.19M
    float* U     = ws + o; o += (size_t)2 * L_SEQ * DINNER;        // 4.19M
    float* XD    = ws + o; o += (size_t)2 * L_SEQ * NXDBL;         // 0.66M
    float* DL    = ws + o; o += (size_t)2 * L_SEQ * DINNER;        // 4.19M
    float* P     = ws + o; o += (size_t)2 * NCHUNK * DINNER * DSTATE;
    float* Hc    = ws + o; o += (size_t)2 * NCHUNK * DINNER * DSTATE;
    float* Hin   = ws + o; o += (size_t)2 * NCHUNK * DINNER * DSTATE;
    float* Y     = ws + o; o += (size_t)2 * L_SEQ * DINNER;        // 4.19M
    float* Ym    = ws + o; o += (size_t)L_SEQ * DMODEL;            // 1.05M
    float* stats = ws + o; o += 16;

    const int T = 256;
    bm_transpose<<<(NBATCH * DMODEL * HW) / T, T, 0, stream>>>(x, xT);
    bm_gemm_xz  <<<64 * 32, T, 0, stream>>>(xT, Win, xz);          // 2048 blocks
    bm_conv     <<<(2 * L_SEQ * DINNER) / T, T, 0, stream>>>(xz, conv_w, conv_b, U);
    bm_gemm_xdbl<<<2 * (L_SEQ / 16) * 3 / 8, T, 0, stream>>>(U, Wx, XD);
    bm_delta    <<<(2 * L_SEQ * DINNER) / T, T, 0, stream>>>(XD, Wdt, bdt, DL);
    bm_scan1    <<<2 * NCHUNK, T, 0, stream>>>(DL, U, XD, A_log, P, Hc);
    bm_scan2    <<<(2 * DINNER * DSTATE) / T, T, 0, stream>>>(P, Hc, Hin);
    bm_scan3    <<<2 * NCHUNK, T, 0, stream>>>(DL, U, XD, A_log, Hin, xz, Dv, Y);
    bm_gemm_out <<<64 * 8, T, 0, stream>>>(Y, Wout, Ym);           // 512 blocks
    bm_gnstats  <<<NBATCH * 4, T, 0, stream>>>(Ym, stats);
    bm_final    <<<(NBATCH * DMODEL * HW) / T, T, 0, stream>>>(Ym, x, gn_w, gn_b, stats, out);
}